// Channel_attention_52578989637795
// MI455X (gfx1250) — compile-verified
//
#include <hip/hip_runtime.h>

typedef __attribute__((ext_vector_type(2))) float v2f;
typedef __attribute__((ext_vector_type(8))) float v8f;

#define N_TOK 4096
#define C_DIM 64

__device__ __forceinline__ v8f wmma_f32_4(v2f a, v2f b, v8f c) {
  // V_WMMA_F32_16X16X4_F32: D = A(16x4) * B(4x16) + C(16x16)
  return __builtin_amdgcn_wmma_f32_16x16x4_f32(
      /*neg_a=*/false, a, /*neg_b=*/false, b,
      /*c_mod=*/(short)0, c, /*reuse_a=*/false, /*reuse_b=*/false);
}

// ---------------------------------------------------------------------------
// Kernel 1: flash attention (softmax(X X^T) X) + elementwise gate by X.
// One wave owns 16 query rows; 8 waves / block -> 128 rows / block.
// Grid = B*N/128 = 128 blocks.
// ---------------------------------------------------------------------------
__global__ __launch_bounds__(256) void attn_gate_kernel(
    const float* __restrict__ x, float* __restrict__ y_ws) {
  __shared__ float lds_p[8 * 256];  // per-wave 16x16 P staging (8 KB)

  const int lane = threadIdx.x & 31;
  const int wave = threadIdx.x >> 5;
  const int col  = lane & 15;   // N (or M for A-matrix) index
  const int half = lane >> 4;   // selects K pair {0,1} vs {2,3}

  const int t0 = blockIdx.x * 128 + wave * 16;  // global token-row base
  const int b  = t0 / N_TOK;
  const int n0 = t0 % N_TOK;
  const float* __restrict__ xb = x + (size_t)b * N_TOK * C_DIM;

  // Q tile in A-layout: qa[k] holds A[m][4k+2*half .. +1], m = col
  const int qrow = n0 + col;
  v2f qa[16];
#pragma unroll
  for (int k = 0; k < 16; ++k)
    qa[k] = *(const v2f*)(xb + (size_t)qrow * C_DIM + 4 * k + 2 * half);

  v8f o[4];
  float rm[8], rl[8];
#pragma unroll
  for (int j = 0; j < 4; ++j) o[j] = (v8f)0.0f;
#pragma unroll
  for (int i = 0; i < 8; ++i) { rm[i] = -1e30f; rl[i] = 0.0f; }

  float* __restrict__ pw = lds_p + wave * 256;

  for (int c0 = 0; c0 < N_TOK; c0 += 16) {
    // ---- S = Q K^T for a 16-column tile (K=64 as 16 chunks of 4) ----
    v8f s = (v8f)0.0f;
#pragma unroll
    for (int k = 0; k < 16; ++k) {
      // B-layout: B[c][n] = X[c0+n][c]; lane holds n=col, K pair by half
      v2f kb = *(const v2f*)(xb + (size_t)(c0 + col) * C_DIM + 4 * k + 2 * half);
      s = wmma_f32_4(qa[k], kb, s);
    }

    // ---- online softmax; C-layout row of element i is (i + 8*half) ----
#pragma unroll
    for (int i = 0; i < 8; ++i) {
      float v = s[i];
#pragma unroll
      for (int off = 1; off < 16; off <<= 1)
        v = fmaxf(v, __shfl_xor(v, off));           // stays within 16-lane half
      float nm = fmaxf(rm[i], v);
      float sc = __expf(rm[i] - nm);
      rm[i] = nm;
      float p = __expf(s[i] - nm);
      s[i] = p;
      float ts = p;
#pragma unroll
      for (int off = 1; off < 16; off <<= 1)
        ts += __shfl_xor(ts, off);
      rl[i] = rl[i] * sc + ts;
#pragma unroll
      for (int j = 0; j < 4; ++j) o[j][i] *= sc;    // rescale accumulators
    }

    // ---- stage P (C-layout -> row-major 16x16 in LDS) ----
#pragma unroll
    for (int i = 0; i < 8; ++i)
      pw[(i + 8 * half) * 16 + col] = s[i];
    asm volatile("" ::: "memory");  // LDS is in-order per wave; just pin issue order

    // ---- reload P in A-layout: pa[kk] = P[m][4kk+2*half .. +1], m = col ----
    v2f pa[4];
#pragma unroll
    for (int kk = 0; kk < 4; ++kk)
      pa[kk] = *(const v2f*)(pw + col * 16 + 4 * kk + 2 * half);
    asm volatile("" ::: "memory");

    // ---- O += P * V  (V = X[c0:c0+16, :]; channels as 4 N-tiles of 16) ----
#pragma unroll
    for (int kk = 0; kk < 4; ++kk) {
      const float* vrow = xb + (size_t)(c0 + 4 * kk + 2 * half) * C_DIM;
#pragma unroll
      for (int j = 0; j < 4; ++j) {
        v2f vb;
        vb.x = vrow[j * 16 + col];           // B[k0  ][n]
        vb.y = vrow[C_DIM + j * 16 + col];   // B[k0+1][n]
        o[j] = wmma_f32_4(pa[kk], vb, o[j]);
      }
    }
  }

  // ---- epilogue: normalize, gate by x, write y to workspace ----
#pragma unroll
  for (int i = 0; i < 8; ++i) {
    float inv = 1.0f / rl[i];
    int row = n0 + i + 8 * half;
#pragma unroll
    for (int j = 0; j < 4; ++j) {
      int ch = j * 16 + col;
      float g = xb[(size_t)row * C_DIM + ch];
      y_ws[((size_t)b * N_TOK + row) * C_DIM + ch] = o[j][i] * inv * g;
    }
  }
}

// ---------------------------------------------------------------------------
// Kernel 2: conv (1,1,4) as GEMM  out[14848,128] = relu(A[14848,256] W[256,128] + b)
// A[m][k] is contiguous in y (k = kw*64+ci), W[k][co] contiguous in conv_w.
// One wave -> 16 rows x 128 cols. 928 tiles / 8 waves = 116 blocks exactly.
// ---------------------------------------------------------------------------
__global__ __launch_bounds__(256) void conv_gemm_kernel(
    const float* __restrict__ y, const float* __restrict__ w,
    const float* __restrict__ bias, float* __restrict__ out) {
  const int lane = threadIdx.x & 31;
  const int wave = threadIdx.x >> 5;
  const int col  = lane & 15;
  const int half = lane >> 4;

  const int rbase = (blockIdx.x * 8 + wave) * 16;  // output-row tile base

  // Per-lane A-matrix row (m = col): decompose row -> (b,d,h,w), w in [0,29)
  int rowg = rbase + col;
  int wpos = rowg % 29;
  int rem  = rowg / 29;
  int h = rem & 31; rem >>= 5;
  int d = rem & 3;
  int bb = rem >> 2;
  size_t abase = (((size_t)bb * N_TOK) + ((d * 32 + h) * 32 + wpos)) * (size_t)C_DIM;

  v8f acc[8];
#pragma unroll
  for (int j = 0; j < 8; ++j) acc[j] = (v8f)0.0f;

  for (int kk = 0; kk < 64; ++kk) {      // K = 256 in chunks of 4
    int k0 = 4 * kk + 2 * half;
    v2f a = *(const v2f*)(y + abase + k0);
    const float* w0 = w + (size_t)k0 * 128;  // weights resident in L2
#pragma unroll
    for (int j = 0; j < 8; ++j) {
      v2f wb;
      wb.x = w0[j * 16 + col];         // W[k0  ][co]
      wb.y = w0[128 + j * 16 + col];   // W[k0+1][co]
      acc[j] = wmma_f32_4(a, wb, acc[j]);
    }
  }

#pragma unroll
  for (int j = 0; j < 8; ++j) {
    int co = j * 16 + col;
    float bv = bias[co];
#pragma unroll
    for (int i = 0; i < 8; ++i) {
      int row = rbase + i + 8 * half;  // C-layout row of element i
      float v = acc[j][i] + bv;
      out[(size_t)row * 128 + co] = fmaxf(v, 0.0f);
    }
  }
}

extern "C" void kernel_launch(void* const* d_in, const int* in_sizes, int n_in,
                              void* d_out, int out_size, void* d_ws, size_t ws_size,
                              hipStream_t stream) {
  const float* x  = (const float*)d_in[0];  // [4,4,32,32,64] f32
  const float* cw = (const float*)d_in[1];  // [1,1,4,64,128] f32 -> [256,128]
  const float* cb = (const float*)d_in[2];  // [128] f32
  float* out = (float*)d_out;               // [4,4,32,29,128] f32
  float* y   = (float*)d_ws;                // scratch: 4*4096*64 f32 = 4 MB

  attn_gate_kernel<<<128, 256, 0, stream>>>(x, y);
  conv_gemm_kernel<<<116, 256, 0, stream>>>(y, cw, cb, out);
}